// LIF_6579889897814
// MI455X (gfx1250) — compile-verified
//
#include <hip/hip_runtime.h>
#include <stdint.h>

// LIF forward scan for MI455X (gfx1250).
// Memory-bound streaming recurrence: async global->LDS DMA pipeline
// (global_load_async_to_lds_b128 + s_wait_asynccnt), wave32, no barriers
// (each thread consumes only the LDS rows it issued -> per-wave ASYNCcnt
// is the only sync needed). Hardware v_tanh_f32 for the sigmoid.

namespace {
constexpr int   Bsz  = 16384;
constexpr int   Lsz  = 2048;
constexpr float kTau = 20.0f;
constexpr float kTh  = 1.0f;
constexpr float kK   = 10.0f;

constexpr int BLOCK = 128;            // 4 wave32s per block
constexpr int CHUNK = 32;             // timesteps per staged chunk = 128 B per lane
constexpr int NC    = Lsz / CHUNK;    // 64 chunks
constexpr int NBUF  = 8;              // LDS ring depth
constexpr int PRE   = NBUF - 1;       // prefetch distance (7 chunks in flight)
constexpr int RSTR  = 36;             // padded row stride in floats (144 B, 16B aligned)

__device__ __forceinline__ float hw_tanhf(float x) {
#if __has_builtin(__builtin_amdgcn_tanhf)
  return __builtin_amdgcn_tanhf(x);   // v_tanh_f32 (TRANS, co-executes with VALU)
#else
  return tanhf(x);
#endif
}

// Copy one 128-byte row chunk global->LDS asynchronously.
// ISA: INST_OFFSET is added to BOTH the LDS dest and the global address,
// so a single base pair + literal offsets covers the whole chunk.
__device__ __forceinline__ void async_load_row_chunk(uint32_t lds_bytes, const float* gp) {
  uint64_t ga = (uint64_t)(uintptr_t)gp;
#define LIF_ALD(OFFSTR)                                                   \
  asm volatile("global_load_async_to_lds_b128 %0, %1, off offset:" OFFSTR \
               :: "v"(lds_bytes), "v"(ga)                                 \
               : "memory")
  LIF_ALD("0");
  LIF_ALD("16");
  LIF_ALD("32");
  LIF_ALD("48");
  LIF_ALD("64");
  LIF_ALD("80");
  LIF_ALD("96");
  LIF_ALD("112");
#undef LIF_ALD
}
}  // namespace

__global__ __launch_bounds__(BLOCK) void lif_scan_async(
    const float* __restrict__ I, float* __restrict__ spikes,
    float* __restrict__ series, float* __restrict__ firstt) {
  __shared__ __align__(16) float tile[NBUF * BLOCK * RSTR];  // 144 KB

  const int tid = threadIdx.x;
  const int row = blockIdx.x * BLOCK + tid;
  const float* gin = I + (size_t)row * Lsz;
  float* gs = spikes + (size_t)row * Lsz;
  float* gn = series + (size_t)row * Lsz;

  // Low 32 bits of the flat shared-aperture address == LDS byte offset.
  const uint32_t lds0 = (uint32_t)(uintptr_t)(&tile[0]);

  // Prologue: fill PRE chunks of the ring (wave ASYNCcnt = 8*PRE = 56).
#pragma unroll
  for (int p = 0; p < PRE; ++p) {
    uint32_t l = lds0 + (uint32_t)(((p * BLOCK) + tid) * RSTR) * 4u;
    async_load_row_chunk(l, gin + p * CHUNK);
  }

  float v = 0.0f, snum = 0.0f;
  int first = Lsz;   // L if never spiked
  int spiked = 0;

  for (int c = 0; c < NC; ++c) {
    if (c < NC - PRE) {
      // Oldest (chunk c) has landed once <= 8*(PRE-1)=48 remain outstanding.
      asm volatile("s_wait_asynccnt 0x30" ::: "memory");
      const int nx = c + PRE;
      const int nb = nx & (NBUF - 1);
      uint32_t l = lds0 + (uint32_t)(((nb * BLOCK) + tid) * RSTR) * 4u;
      async_load_row_chunk(l, gin + nx * CHUNK);   // peak ASYNCcnt = 56 <= 63
    } else {
      // Tail: no more issues, drain everything (over-waits only last 7 chunks).
      asm volatile("s_wait_asynccnt 0x0" ::: "memory");
    }

    const int b = c & (NBUF - 1);
    const float* rt = &tile[((b * BLOCK) + tid) * RSTR];
    const int t0 = c * CHUNK;

#pragma unroll
    for (int q4 = 0; q4 < CHUNK / 4; ++q4) {
      const float4 x4 = *(const float4*)(rt + q4 * 4);   // ds_load_b128
      float xs[4] = {x4.x, x4.y, x4.z, x4.w};
      float so[4], no[4];
#pragma unroll
      for (int q = 0; q < 4; ++q) {
        const float x = xs[q];
        // Euler update: v = v + (-v/tau + I_t)
        v = v + (v * (-1.0f / kTau) + x);
        // XLA sigmoid(z) = 0.5*tanh(0.5*z) + 0.5
        const float z     = kK * (v - kTh);
        const float ssoft = 0.5f * hw_tanhf(0.5f * z) + 0.5f;
        const float shard = (v >= kTh) ? 1.0f : 0.0f;
        // STE forward kept literally: not exactly 0/1 in fp32; the reset
        // v*(1-s) depends on this rounding, so reproduce it.
        const float s = (shard - ssoft) + ssoft;
        snum += s;
        if (!spiked && shard > 0.0f) { first = t0 + q4 * 4 + q; spiked = 1; }
        v = v * (1.0f - s);
        so[q] = s;
        no[q] = snum;
      }
      *(float4*)(gs + t0 + q4 * 4) = make_float4(so[0], so[1], so[2], so[3]);
      *(float4*)(gn + t0 + q4 * 4) = make_float4(no[0], no[1], no[2], no[3]);
    }
  }
  firstt[row] = (float)first;
}

extern "C" void kernel_launch(void* const* d_in, const int* in_sizes, int n_in,
                              void* d_out, int out_size, void* d_ws, size_t ws_size,
                              hipStream_t stream) {
  (void)in_sizes; (void)n_in; (void)out_size; (void)d_ws; (void)ws_size;
  const float* I = (const float*)d_in[0];
  float* out = (float*)d_out;
  float* spikes = out;                                   // [B, L]
  float* series = out + (size_t)Bsz * Lsz;               // [B, L]
  float* firstt = out + 2ull * (size_t)Bsz * Lsz;        // [B] (as float)

  dim3 grid(Bsz / BLOCK);
  dim3 block(BLOCK);
  lif_scan_async<<<grid, block, 0, stream>>>(I, spikes, series, firstt);
}